// PowerSpectrum_51127290691590
// MI455X (gfx1250) — compile-verified
//
#include <hip/hip_runtime.h>

typedef __attribute__((ext_vector_type(2))) float v2f;
typedef __attribute__((ext_vector_type(8))) float v8f;

#define L_DIM 4
#define M_DIM 7
#define FNU   32
#define F1    16
#define NPW   4   // n-samples per wave

// out[n, l*512 + q*16 + p] = cg[l] * sum_m x_nu[l,n,m,q] * x_1[l,n,m,p]
__global__ __launch_bounds__(256) void ps_wmma_kernel(
    const float* __restrict__ x_nu,   // [L, N, M, FNU]
    const float* __restrict__ x_1,    // [L, N, M, F1]
    float* __restrict__ out,          // [N, L*FNU*F1]
    int N)
{
    const int tid  = blockIdx.x * blockDim.x + threadIdx.x;
    const int wave = tid >> 5;
    const int lane = tid & 31;
    const int half = lane >> 4;   // 0 = lanes 0-15, 1 = lanes 16-31
    const int lid  = lane & 15;

    const int waves_per_l = (N + NPW - 1) / NPW;   // 12500 for N=50000
    const int l  = wave / waves_per_l;             // wave-uniform
    if (l >= L_DIM) return;                        // uniform branch, EXEC stays full
    const int n0 = (wave % waves_per_l) * NPW;

    const float cg = __frsqrt_rn(2.0f * (float)l + 1.0f);

    const float* xnu_l = x_nu + (size_t)l * N * (M_DIM * FNU);
    const float* x1_l  = x_1  + (size_t)l * N * (M_DIM * F1);

    // K-step m-row indices per lane half (A layout: lanes 0-15 -> K=0,1 ; 16-31 -> K=2,3)
    const int mLo0 = 2 * half;           // kstep0: rows {mLo0, mLo0+1}  => m = 0..3
    const int mHi0 = 4 + 2 * half;       // kstep1: rows {mHi0, mHi0+1}  => m = 4..7 (7 = pad)
    const int mHi1 = (half == 0) ? (mHi0 + 1) : 6;   // clamp pad row to valid memory
    const float wHi = (half == 0) ? 1.0f : 0.0f;     // zero weight for pad row m=7

    for (int i = 0; i < NPW; ++i) {
        const int n = n0 + i;
        if (n >= N) break;               // wave-uniform

        const float* xa = xnu_l + (size_t)n * (M_DIM * FNU);   // [m][q]
        const float* xb = x1_l  + (size_t)n * (M_DIM * F1);    // [m][p]
        float* op = out + (size_t)n * (L_DIM * FNU * F1) + l * (FNU * F1);

        if (i + 1 < NPW) {               // warm next sample's inputs (speculative)
            __builtin_prefetch(xa + M_DIM * FNU + lane, 0, 1);
            __builtin_prefetch(xb + M_DIM * F1 + lane, 0, 1);
        }

        // B fragments (4x16 = m x p), cg folded in
        v2f b0, b1;
        b0.x = cg * __builtin_nontemporal_load(xb + mLo0 * F1 + lid);
        b0.y = cg * __builtin_nontemporal_load(xb + (mLo0 + 1) * F1 + lid);
        b1.x = cg * __builtin_nontemporal_load(xb + mHi0 * F1 + lid);
        b1.y = cg * wHi * __builtin_nontemporal_load(xb + mHi1 * F1 + lid);

        // A fragments: tile0 -> q = lid, tile1 -> q = 16 + lid   (16x4 = q x m)
        v2f a00, a01, a10, a11;
        a00.x = __builtin_nontemporal_load(xa + mLo0 * FNU + lid);
        a00.y = __builtin_nontemporal_load(xa + (mLo0 + 1) * FNU + lid);
        a01.x = __builtin_nontemporal_load(xa + mHi0 * FNU + lid);
        a01.y = wHi * __builtin_nontemporal_load(xa + mHi1 * FNU + lid);

        a10.x = __builtin_nontemporal_load(xa + mLo0 * FNU + 16 + lid);
        a10.y = __builtin_nontemporal_load(xa + (mLo0 + 1) * FNU + 16 + lid);
        a11.x = __builtin_nontemporal_load(xa + mHi0 * FNU + 16 + lid);
        a11.y = wHi * __builtin_nontemporal_load(xa + mHi1 * FNU + 16 + lid);

        v8f c0 = {0.f,0.f,0.f,0.f,0.f,0.f,0.f,0.f};
        v8f c1 = {0.f,0.f,0.f,0.f,0.f,0.f,0.f,0.f};

        // D = A*B + C ; two K=4 steps cover m=0..7 (m=7 zero-padded)
        c0 = __builtin_amdgcn_wmma_f32_16x16x4_f32(false, a00, false, b0, (short)0, c0, false, false);
        c0 = __builtin_amdgcn_wmma_f32_16x16x4_f32(false, a01, false, b1, (short)0, c0, false, false);
        c1 = __builtin_amdgcn_wmma_f32_16x16x4_f32(false, a10, false, b0, (short)0, c1, false, false);
        c1 = __builtin_amdgcn_wmma_f32_16x16x4_f32(false, a11, false, b1, (short)0, c1, false, false);

        // C/D layout: VGPR r holds row M=r (lanes 0-15) / M=r+8 (lanes 16-31), col N=lid
        float* o0 = op + (size_t)(8 * half) * F1 + lid;          // tile0: q = r + 8*half
        float* o1 = op + (size_t)(16 + 8 * half) * F1 + lid;     // tile1: q = 16 + r + 8*half
#pragma unroll
        for (int r = 0; r < 8; ++r) {
            __builtin_nontemporal_store(c0[r], o0 + r * F1);
            __builtin_nontemporal_store(c1[r], o1 + r * F1);
        }
    }
}

extern "C" void kernel_launch(void* const* d_in, const int* in_sizes, int n_in,
                              void* d_out, int out_size, void* d_ws, size_t ws_size,
                              hipStream_t stream) {
    const float* x_nu = (const float*)d_in[0];
    const float* x_1  = (const float*)d_in[1];
    float* out = (float*)d_out;

    const int N = in_sizes[0] / (L_DIM * M_DIM * FNU);   // 50000

    const int waves_per_l = (N + NPW - 1) / NPW;
    const long total_waves = (long)L_DIM * waves_per_l;          // 50000
    const long total_threads = total_waves * 32;
    const int block = 256;
    const int grid = (int)((total_threads + block - 1) / block); // 6250

    hipLaunchKernelGGL(ps_wmma_kernel, dim3(grid), dim3(block), 0, stream,
                       x_nu, x_1, out, N);
}